// HGT_44203803411105
// MI455X (gfx1250) — compile-verified
//
#include <hip/hip_runtime.h>
#include <math.h>

typedef __attribute__((ext_vector_type(2))) float v2f;
typedef __attribute__((ext_vector_type(8))) float v8f;

#define C_DIM   128
#define H_DIM   4
#define D_DIM   32
#define NUSER   100000
#define NNEWS   20000
#define E_EDGES 250000

// ---------- helpers ----------
__device__ __forceinline__ unsigned flip_f(float f) {
  unsigned u = __float_as_uint(f);
  return (u & 0x80000000u) ? ~u : (u | 0x80000000u);   // monotone map for atomicMax
}
__device__ __forceinline__ float unflip_f(unsigned u) {
  u = (u & 0x80000000u) ? (u & 0x7fffffffu) : ~u;
  return __uint_as_float(u);
}
__device__ __forceinline__ float gelu_f(float x) {
  return 0.5f * x * (1.0f + erff(x * 0.70710678118654752f));
}

// CDNA5 async global->LDS copy (16B per lane), tracked by ASYNCcnt.
// Generic pointer to __shared__ carries the LDS byte offset in its low 32 bits
// (flat aperture rule, CDNA5 ISA).
__device__ __forceinline__ void async_cp16(const float* g, const float* l) {
  unsigned loff = (unsigned)(unsigned long long)(const void*)l;
  asm volatile("global_load_async_to_lds_b128 %0, %1, off"
               :: "v"(loff), "v"(g) : "memory");
}
__device__ __forceinline__ void wait_async0() {
  asm volatile("s_wait_asynccnt 0x0" ::: "memory");
}

// ---------- zero fill ----------
__global__ __launch_bounds__(256) void hgt_zero(float4* __restrict__ p, int n4) {
  int i = blockIdx.x * blockDim.x + threadIdx.x;
  if (i < n4) p[i] = make_float4(0.f, 0.f, 0.f, 0.f);
}

// ---------- K/Q/V projection: OUT[n,c] = sum_k X[n,k]*W[k,c] + b[c] ----------
// grid.x = N/16 (row tiles); 8 waves = 8 column tiles. Weights pipelined
// through LDS in 16-row strips via async copies, double-buffered.
__global__ __launch_bounds__(256) void hgt_proj_qkv(
    const float* __restrict__ X,
    const float* __restrict__ Wk, const float* __restrict__ bk,
    const float* __restrict__ Wq, const float* __restrict__ bq,
    const float* __restrict__ Wv, const float* __restrict__ bv,
    float* __restrict__ Kout, float* __restrict__ Qout, float* __restrict__ Vout) {
  __shared__ float wl[2][3][16][C_DIM];   // 48 KB
  const int tid = threadIdx.x;
  const int wave = tid >> 5;
  const int lane = tid & 31;
  const int lm = lane & 15;
  const int lh = lane >> 4;
  const int row0 = blockIdx.x * 16;
  const int col0 = wave * 16;
  const int cc = col0 + lm;

  const float* Ws[3] = {Wk, Wq, Wv};
  auto stage = [&](int s, int b) {
#pragma unroll
    for (int i = 0; i < 6; ++i) {
      const int idx = i * 256 + tid;       // 1536 float4 per strip
      const int mat = idx >> 9;
      const int rem = idx & 511;
      const int row = rem >> 5;
      const int c4  = rem & 31;
      async_cp16(Ws[mat] + (size_t)(s * 16 + row) * C_DIM + c4 * 4,
                 &wl[b][mat][row][c4 * 4]);
    }
  };

  v8f ck = {}, cq = {}, cv = {};
  const float* arow = X + (size_t)(row0 + lm) * C_DIM + 2 * lh;

  stage(0, 0);
  for (int s = 0; s < 8; ++s) {
    wait_async0();
    __syncthreads();
    if (s + 1 < 8) stage(s + 1, (s + 1) & 1);
    const int b = s & 1;
#pragma unroll
    for (int kk = 0; kk < 16; kk += 4) {
      const int kr = kk + 2 * lh;
      v2f a = *(const v2f*)(arow + s * 16 + kk);
      v2f bkf, bqf, bvf;
      bkf.x = wl[b][0][kr][cc];     bkf.y = wl[b][0][kr + 1][cc];
      bqf.x = wl[b][1][kr][cc];     bqf.y = wl[b][1][kr + 1][cc];
      bvf.x = wl[b][2][kr][cc];     bvf.y = wl[b][2][kr + 1][cc];
      ck = __builtin_amdgcn_wmma_f32_16x16x4_f32(false, a, false, bkf, (short)0, ck, false, false);
      cq = __builtin_amdgcn_wmma_f32_16x16x4_f32(false, a, false, bqf, (short)0, cq, false, false);
      cv = __builtin_amdgcn_wmma_f32_16x16x4_f32(false, a, false, bvf, (short)0, cv, false, false);
    }
  }
#pragma unroll
  for (int j = 0; j < 8; ++j) {
    const int r = row0 + j + 8 * lh;
    Kout[(size_t)r * C_DIM + cc] = ck[j] + bk[cc];
    Qout[(size_t)r * C_DIM + cc] = cq[j] + bq[cc];
    Vout[(size_t)r * C_DIM + cc] = cv[j] + bv[cc];
  }
}

// ---------- relation transform: krel[n,h,e] = sum_d k[n,h,d]*A[h,d,e]; same for v/M ----------
// grid.x = N/16; wave = (head, col-tile). a_rel/m_rel async-staged to LDS once.
__global__ __launch_bounds__(256) void hgt_rel(
    const float* __restrict__ Kin, const float* __restrict__ Vin,
    const float* __restrict__ Arel, const float* __restrict__ Mrel,
    float* __restrict__ Krel, float* __restrict__ Vrel) {
  __shared__ float al[H_DIM][D_DIM][D_DIM];  // 16 KB
  __shared__ float ml[H_DIM][D_DIM][D_DIM];  // 16 KB
  const int tid = threadIdx.x;
  const int wave = tid >> 5;
  const int lane = tid & 31;
  const int lm = lane & 15;
  const int lh = lane >> 4;
  const int h  = wave >> 1;
  const int col0 = (wave & 1) * 16;
  const int row0 = blockIdx.x * 16;
  const int cc = col0 + lm;

#pragma unroll
  for (int i = 0; i < 4; ++i) {
    const int idx = (i * 256 + tid) * 4;   // float index, 4096 per matrix
    async_cp16(Arel + idx, &al[0][0][0] + idx);
    async_cp16(Mrel + idx, &ml[0][0][0] + idx);
  }
  wait_async0();
  __syncthreads();

  v8f ckr = {}, cvr = {};
  const float* krow = Kin + (size_t)(row0 + lm) * C_DIM + h * D_DIM + 2 * lh;
  const float* vrow = Vin + (size_t)(row0 + lm) * C_DIM + h * D_DIM + 2 * lh;
#pragma unroll
  for (int kk = 0; kk < D_DIM; kk += 4) {
    const int kr = kk + 2 * lh;
    v2f ak = *(const v2f*)(krow + kk);
    v2f av = *(const v2f*)(vrow + kk);
    v2f bA, bM;
    bA.x = al[h][kr][cc];     bA.y = al[h][kr + 1][cc];
    bM.x = ml[h][kr][cc];     bM.y = ml[h][kr + 1][cc];
    ckr = __builtin_amdgcn_wmma_f32_16x16x4_f32(false, ak, false, bA, (short)0, ckr, false, false);
    cvr = __builtin_amdgcn_wmma_f32_16x16x4_f32(false, av, false, bM, (short)0, cvr, false, false);
  }
#pragma unroll
  for (int j = 0; j < 8; ++j) {
    const int r = row0 + j + 8 * lh;
    const int c = h * D_DIM + cc;
    Krel[(size_t)r * C_DIM + c] = ckr[j];
    Vrel[(size_t)r * C_DIM + c] = cvr[j];
  }
}

// ---------- edge score + segment max ----------
__global__ __launch_bounds__(256) void hgt_edge_score(
    const int* __restrict__ src, const int* __restrict__ dst,
    const float* __restrict__ Qdt, const float* __restrict__ Krel,
    const float* __restrict__ prel,
    float* __restrict__ scores, unsigned* __restrict__ mbuf) {
  int idx = blockIdx.x * blockDim.x + threadIdx.x;
  if (idx >= E_EDGES * H_DIM) return;
  const int e = idx >> 2, h = idx & 3;
  const int s = src[e], d = dst[e];
  const float4* qr = (const float4*)(Qdt  + (size_t)d * C_DIM + h * D_DIM);
  const float4* kr = (const float4*)(Krel + (size_t)s * C_DIM + h * D_DIM);
  float acc = 0.f;
#pragma unroll
  for (int i = 0; i < 8; ++i) {
    float4 a = qr[i], b = kr[i];
    acc += a.x * b.x + a.y * b.y + a.z * b.z + a.w * b.w;
  }
  const float sc = acc * prel[h] * 0.17677669529663688f; // 1/sqrt(32)
  scores[idx] = sc;
  atomicMax(&mbuf[(size_t)d * H_DIM + h], flip_f(sc));
}

// ---------- exp(score - max) + segment sum ----------
__global__ __launch_bounds__(256) void hgt_edge_exp(
    const int* __restrict__ dst,
    float* __restrict__ scores, const unsigned* __restrict__ mbuf,
    float* __restrict__ den) {
  int idx = blockIdx.x * blockDim.x + threadIdx.x;
  if (idx >= E_EDGES * H_DIM) return;
  const int e = idx >> 2, h = idx & 3;
  const int d = dst[e];
  const float m = unflip_f(mbuf[(size_t)d * H_DIM + h]);
  const float ex = __expf(scores[idx] - m);
  scores[idx] = ex;
  atomicAdd(&den[(size_t)d * H_DIM + h], ex);
}

// ---------- normalize + scatter-add messages ----------
__global__ __launch_bounds__(256) void hgt_edge_scatter(
    const int* __restrict__ src, const int* __restrict__ dst,
    const float* __restrict__ scores, const float* __restrict__ den,
    const float* __restrict__ Vrel, float* __restrict__ agg) {
  long long idx = (long long)blockIdx.x * blockDim.x + threadIdx.x;
  if (idx >= (long long)E_EDGES * C_DIM) return;
  const int e = (int)(idx >> 7);
  const int c = (int)(idx & 127);
  const int h = c >> 5;
  const int s = src[e], d = dst[e];
  const float ex = scores[(size_t)e * H_DIM + h];
  const float dn = den[(size_t)d * H_DIM + h];
  const float alpha = ex / (dn + 1e-16f);
  atomicAdd(&agg[(size_t)d * C_DIM + c], Vrel[(size_t)s * C_DIM + c] * alpha);
}

// ---------- output: relu( a*(gelu(agg)@Wa + ba) + (1-a)*x_old ) ----------
__global__ __launch_bounds__(256) void hgt_out(
    const float* __restrict__ Agg, const float* __restrict__ Wa,
    const float* __restrict__ ba, const float* __restrict__ skip,
    const float* __restrict__ Xold, float* __restrict__ Xnew) {
  __shared__ float wl[2][16][C_DIM];   // 16 KB
  const int tid = threadIdx.x;
  const int wave = tid >> 5;
  const int lane = tid & 31;
  const int lm = lane & 15;
  const int lh = lane >> 4;
  const int row0 = blockIdx.x * 16;
  const int col0 = wave * 16;
  const int cc = col0 + lm;

  auto stage = [&](int s, int b) {
#pragma unroll
    for (int i = 0; i < 2; ++i) {
      const int idx = i * 256 + tid;       // 512 float4 per strip
      const int row = idx >> 5;
      const int c4  = idx & 31;
      async_cp16(Wa + (size_t)(s * 16 + row) * C_DIM + c4 * 4,
                 &wl[b][row][c4 * 4]);
    }
  };

  v8f acc = {};
  const float* arow = Agg + (size_t)(row0 + lm) * C_DIM + 2 * lh;

  stage(0, 0);
  for (int s = 0; s < 8; ++s) {
    wait_async0();
    __syncthreads();
    if (s + 1 < 8) stage(s + 1, (s + 1) & 1);
    const int b = s & 1;
#pragma unroll
    for (int kk = 0; kk < 16; kk += 4) {
      const int kr = kk + 2 * lh;
      v2f a = *(const v2f*)(arow + s * 16 + kk);
      a.x = gelu_f(a.x);
      a.y = gelu_f(a.y);
      v2f bw;
      bw.x = wl[b][kr][cc];
      bw.y = wl[b][kr + 1][cc];
      acc = __builtin_amdgcn_wmma_f32_16x16x4_f32(false, a, false, bw, (short)0, acc, false, false);
    }
  }
  const float al = 1.f / (1.f + __expf(-skip[0]));
#pragma unroll
  for (int j = 0; j < 8; ++j) {
    const int r = row0 + j + 8 * lh;
    const float o = acc[j] + ba[cc];
    const float blend = al * o + (1.f - al) * Xold[(size_t)r * C_DIM + cc];
    Xnew[(size_t)r * C_DIM + cc] = fmaxf(blend, 0.f);
  }
}

// ---------- orchestration ----------
extern "C" void kernel_launch(void* const* d_in, const int* in_sizes, int n_in,
                              void* d_out, int out_size, void* d_ws, size_t ws_size,
                              hipStream_t stream) {
  (void)in_sizes; (void)n_in; (void)out_size; (void)ws_size;
  const float* x_user = (const float*)d_in[0];
  const float* x_news = (const float*)d_in[1];
  const int* ei[3] = {(const int*)d_in[2], (const int*)d_in[3], (const int*)d_in[4]};
  const float* Wk = (const float*)d_in[5];
  const float* bk = (const float*)d_in[6];
  const float* Wq = (const float*)d_in[7];
  const float* bq = (const float*)d_in[8];
  const float* Wv = (const float*)d_in[9];
  const float* bv = (const float*)d_in[10];
  const float* Wa = (const float*)d_in[11];
  const float* ba = (const float*)d_in[12];
  const float* skip = (const float*)d_in[13];
  const float* a_rel = (const float*)d_in[14];
  const float* m_rel = (const float*)d_in[15];
  const float* p_rel = (const float*)d_in[16];
  float* out = (float*)d_out;

  const size_t NU = (size_t)NUSER * C_DIM;
  const size_t NN = (size_t)NNEWS * C_DIM;

  float* p = (float*)d_ws;
  float* xu_a = p; p += NU;
  float* xn_a = p; p += NN;
  float* kb[2]; float* qb[2]; float* vb[2]; float* ag[2];
  kb[0] = p; p += NU;  qb[0] = p; p += NU;  vb[0] = p; p += NU;
  kb[1] = p; p += NN;  qb[1] = p; p += NN;  vb[1] = p; p += NN;
  float* krel = p; p += NU;
  float* vrel = p; p += NU;
  ag[0] = p; p += NU;
  ag[1] = p; p += NN;
  float* scores = p; p += (size_t)E_EDGES * H_DIM;
  float* den = p; p += (size_t)NUSER * H_DIM;
  unsigned* mbuf = (unsigned*)p;

  const int sizes[2] = {NUSER, NNEWS};
  const int est[3] = {0, 1, 0};
  const int edt[3] = {1, 0, 0};

  const int eh_blocks = (E_EDGES * H_DIM + 255) / 256;
  const int ec_blocks = (int)(((long long)E_EDGES * C_DIM + 255) / 256);

  for (int l = 0; l < 2; ++l) {
    const float* xin[2]  = { l ? xu_a : x_user, l ? xn_a : x_news };
    float* xout[2] = { l ? out : xu_a, l ? out + NU : xn_a };

    // K/Q/V projections per node type (WMMA f32 + async LDS weight pipeline)
    for (int t = 0; t < 2; ++t) {
      const size_t wo = ((size_t)l * 2 + t) * C_DIM * C_DIM;
      const size_t bo = ((size_t)l * 2 + t) * C_DIM;
      hgt_proj_qkv<<<sizes[t] / 16, 256, 0, stream>>>(
          xin[t], Wk + wo, bk + bo, Wq + wo, bq + bo, Wv + wo, bv + bo,
          kb[t], qb[t], vb[t]);
    }

    // zero aggregation buffers
    hgt_zero<<<(int)(NU / 4 + 255) / 256, 256, 0, stream>>>((float4*)ag[0], (int)(NU / 4));
    hgt_zero<<<(int)(NN / 4 + 255) / 256, 256, 0, stream>>>((float4*)ag[1], (int)(NN / 4));

    for (int r = 0; r < 3; ++r) {
      const int st = est[r], dt = edt[r];
      const size_t ro = ((size_t)l * 3 + r) * H_DIM * D_DIM * D_DIM;
      const float* pr = p_rel + ((size_t)l * 3 + r) * H_DIM;

      // relation-specific K/V transforms (WMMA f32, per head)
      hgt_rel<<<sizes[st] / 16, 256, 0, stream>>>(
          kb[st], vb[st], a_rel + ro, m_rel + ro, krel, vrel);

      // reset segment-max / denominator for destination type
      const int nseg4 = sizes[dt] * H_DIM / 4;
      hgt_zero<<<(nseg4 + 255) / 256, 256, 0, stream>>>((float4*)den, nseg4);
      hgt_zero<<<(nseg4 + 255) / 256, 256, 0, stream>>>((float4*)mbuf, nseg4);

      hgt_edge_score<<<eh_blocks, 256, 0, stream>>>(
          ei[r], ei[r] + E_EDGES, qb[dt], krel, pr, scores, mbuf);
      hgt_edge_exp<<<eh_blocks, 256, 0, stream>>>(
          ei[r] + E_EDGES, scores, mbuf, den);
      hgt_edge_scatter<<<ec_blocks, 256, 0, stream>>>(
          ei[r], ei[r] + E_EDGES, scores, den, vrel, ag[dt]);
    }

    // output projection + skip + relu per node type
    for (int t = 0; t < 2; ++t) {
      const size_t wo = ((size_t)l * 2 + t) * C_DIM * C_DIM;
      const size_t bo = ((size_t)l * 2 + t) * C_DIM;
      hgt_out<<<sizes[t] / 16, 256, 0, stream>>>(
          ag[t], Wa + wo, ba + bo, skip + (size_t)l * 2 + t, xin[t], xout[t]);
    }
  }
}